// GatedGraphConv_7782480740942
// MI455X (gfx1250) — compile-verified
//
#include <hip/hip_runtime.h>
#include <math.h>

// GatedGraphConv for MI455X (gfx1250, wave32).
// Strategy: materialize per-edge W_e = efeat @ W_edge + b_edge ONCE via
// V_WMMA_F32_16X16X4_F32 (f32-exact), then 3x { zero agg; per-edge matvec +
// atomic scatter; GRU cell }.

typedef __attribute__((ext_vector_type(2))) float v2f;
typedef __attribute__((ext_vector_type(8))) float v8f;

#define Dq   32
#define EFq  64
#define DDq  1024   // Dq*Dq

// ---------------------------------------------------------------------------
// K1: W_e[e, d*32+o] = sum_f efeat[e,f] * W_edge[f, d*32+o] + b_edge[d*32+o]
// One wave handles a 32-edge super-tile = two 16x1024 WMMA strips sharing the
// B fragments (halves W_edge/L2 traffic).  K=64 -> 16 slabs of K=4.
// Fragment layouts per CDNA5 ISA 7.12.2:
//   A 16x4 f32 : lane l -> M=l&15, VGPR pair = K {2*(l>>4), 2*(l>>4)+1} of slab
//   B 4x16 f32 : lane l -> N=l&15, VGPR pair = K {2*(l>>4), 2*(l>>4)+1}
//   C 16x16 f32: VGPR v, lane l -> M = v + 8*(l>>4), N = l&15
// ---------------------------------------------------------------------------
__global__ __launch_bounds__(64) void edge_gemm_wmma(
    const float* __restrict__ efeat, const float* __restrict__ W_edge,
    const float* __restrict__ b_edge, float* __restrict__ W_e, int E)
{
  const int lane = threadIdx.x & 31;
  const int wv   = threadIdx.x >> 5;
  const int tp   = blockIdx.x * 2 + wv;     // 32-edge super-tile index
  const int e0   = tp * 32;
  if (e0 + 32 > E) return;                  // E is a multiple of 32

  const int r  = lane & 15;
  const int hh = lane >> 4;

  // Preload A fragments for both 16-edge tiles (each element read exactly once)
  v2f a0[16], a1[16];
  const float* eA = efeat + (size_t)(e0 + r) * EFq;
  const float* eB = efeat + (size_t)(e0 + 16 + r) * EFq;
#pragma unroll
  for (int s = 0; s < 16; ++s) {
    const int k = 4 * s + 2 * hh;
    a0[s] = *(const v2f*)(eA + k);
    a1[s] = *(const v2f*)(eB + k);
  }

  for (int nn = 0; nn < 64; ++nn) {         // 64 column tiles of 16
    const int c = nn * 16 + r;              // this lane's output column
    v8f acc0 = {};
    v8f acc1 = {};
#pragma unroll
    for (int s = 0; s < 16; ++s) {
      const int kk = 4 * s + 2 * hh;
      v2f b;
      b.x = W_edge[kk * DDq + c];
      b.y = W_edge[(kk + 1) * DDq + c];
      acc0 = __builtin_amdgcn_wmma_f32_16x16x4_f32(
          false, a0[s], false, b, (short)0, acc0, false, false);
      acc1 = __builtin_amdgcn_wmma_f32_16x16x4_f32(
          false, a1[s], false, b, (short)0, acc1, false, false);
    }
    const float bias = b_edge[c];
#pragma unroll
    for (int v = 0; v < 8; ++v) {
      const int row0 = e0 + v + 8 * hh;          // C-matrix M index, tile 0
      const int row1 = e0 + 16 + v + 8 * hh;     // tile 1
      W_e[(size_t)row0 * DDq + c] = acc0[v] + bias;
      W_e[(size_t)row1 * DDq + c] = acc1[v] + bias;
    }
  }
}

// ---------------------------------------------------------------------------
// K2: zero fill (agg buffer) -- workspace is poisoned, must self-initialize.
// ---------------------------------------------------------------------------
__global__ __launch_bounds__(256) void zero_f4(float4* __restrict__ p, int n4)
{
  int i = blockIdx.x * 256 + threadIdx.x;
  if (i < n4) p[i] = make_float4(0.f, 0.f, 0.f, 0.f);
}

// ---------------------------------------------------------------------------
// K3: msg[e,:] = h[src[e],:] @ W_e[e]  ;  agg[dst[e],:] += msg  (atomics)
// Wave per edge; lane = output column o.  W_e reads are fully coalesced
// (128B per d); h row gather is one coalesced 128B row.
// ---------------------------------------------------------------------------
__global__ __launch_bounds__(256) void message_scatter(
    const float* __restrict__ h, const float* __restrict__ W_e,
    const int* __restrict__ src, const int* __restrict__ dst,
    float* __restrict__ agg, int E)
{
  const int e = blockIdx.x * 8 + (threadIdx.x >> 5);
  if (e >= E) return;
  const int l = threadIdx.x & 31;
  const int s = src[e];
  const int t = dst[e];
  const float hv = h[(size_t)s * Dq + l];
  const float* We = W_e + (size_t)e * DDq;
  float m = 0.f;
#pragma unroll
  for (int d = 0; d < 32; ++d)
    m = fmaf(__shfl(hv, d, 32), We[d * 32 + l], m);
  atomicAdd(&agg[(size_t)t * Dq + l], m);
}

// ---------------------------------------------------------------------------
// K4: GRU cell, wave per node, lane = output unit o.
//   gi = x@W_ih^T + b_ih ; gh = h@W_hh^T + b_hh (rows ordered r,z,n)
//   r=sig(i_r+h_r) z=sig(i_z+h_z) n=tanh(i_n + r*h_n) ; h' = (1-z)*n + z*h
// ---------------------------------------------------------------------------
__device__ __forceinline__ float sigmoidf_(float x) {
  return 1.f / (1.f + expf(-x));
}

__global__ __launch_bounds__(256) void gru_update(
    const float* __restrict__ a, const float* __restrict__ h,
    const float* __restrict__ W_ih, const float* __restrict__ b_ih,
    const float* __restrict__ W_hh, const float* __restrict__ b_hh,
    float* __restrict__ hout, int N)
{
  const int n = blockIdx.x * 8 + (threadIdx.x >> 5);
  if (n >= N) return;
  const int o = threadIdx.x & 31;

  const float xv = a[(size_t)n * Dq + o];
  const float hv = h[(size_t)n * Dq + o];

  float gr = b_ih[o],      hr = b_hh[o];
  float gz = b_ih[32 + o], hz = b_hh[32 + o];
  float gn = b_ih[64 + o], hn = b_hh[64 + o];

  const float* wiR = W_ih + o * Dq;
  const float* wiZ = W_ih + (32 + o) * Dq;
  const float* wiN = W_ih + (64 + o) * Dq;
  const float* whR = W_hh + o * Dq;
  const float* whZ = W_hh + (32 + o) * Dq;
  const float* whN = W_hh + (64 + o) * Dq;

#pragma unroll
  for (int d = 0; d < 32; ++d) {
    const float xd = __shfl(xv, d, 32);   // wave32 broadcast
    const float hd = __shfl(hv, d, 32);
    gr = fmaf(wiR[d], xd, gr);
    gz = fmaf(wiZ[d], xd, gz);
    gn = fmaf(wiN[d], xd, gn);
    hr = fmaf(whR[d], hd, hr);
    hz = fmaf(whZ[d], hd, hz);
    hn = fmaf(whN[d], hd, hn);
  }
  const float rr = sigmoidf_(gr + hr);
  const float zz = sigmoidf_(gz + hz);
  const float nnv = tanhf(gn + rr * hn);
  hout[(size_t)n * Dq + o] = (1.f - zz) * nnv + zz * hv;
}

// ---------------------------------------------------------------------------
// Host side
// ---------------------------------------------------------------------------
extern "C" void kernel_launch(void* const* d_in, const int* in_sizes, int n_in,
                              void* d_out, int out_size, void* d_ws, size_t ws_size,
                              hipStream_t stream) {
  const float* feat   = (const float*)d_in[0];
  const float* efeat  = (const float*)d_in[1];
  const float* W_edge = (const float*)d_in[2];
  const float* b_edge = (const float*)d_in[3];
  const float* W_ih   = (const float*)d_in[4];
  const float* b_ih   = (const float*)d_in[5];
  const float* W_hh   = (const float*)d_in[6];
  const float* b_hh   = (const float*)d_in[7];
  const int*   src    = (const int*)d_in[8];
  const int*   dst    = (const int*)d_in[9];

  const int E = in_sizes[8];
  const int N = in_sizes[0] / Dq;

  // Workspace layout: [ W_e : E*1024 f32 ][ agg : N*32 ][ hA : N*32 ][ hB : N*32 ]
  float* W_e = (float*)d_ws;
  float* agg = W_e + (size_t)E * DDq;
  float* hA  = agg + (size_t)N * Dq;
  float* hB  = hA + (size_t)N * Dq;

  // 1. Per-edge weight matrices via WMMA (constant across steps -> compute once)
  {
    const int superTiles = E / 32;                 // E divisible by 32
    const int blocks = (superTiles + 1) / 2;       // 2 waves per block
    edge_gemm_wmma<<<blocks, 64, 0, stream>>>(efeat, W_edge, b_edge, W_e, E);
  }

  const int n4     = (N * Dq) / 4;
  const int zb     = (n4 + 255) / 256;
  const int msgb   = (E + 7) / 8;
  const int grub   = (N + 7) / 8;

  const float* hin = feat;
  for (int step = 0; step < 3; ++step) {
    float* hout = (step == 2) ? (float*)d_out : ((step == 0) ? hA : hB);
    zero_f4<<<zb, 256, 0, stream>>>((float4*)agg, n4);
    message_scatter<<<msgb, 256, 0, stream>>>(hin, W_e, src, dst, agg, E);
    gru_update<<<grub, 256, 0, stream>>>(agg, hin, W_ih, b_ih, W_hh, b_hh, hout, N);
    hin = hout;
  }
}